// InventoryAwareLogicNet_5471788335895
// MI455X (gfx1250) — compile-verified
//
#include <hip/hip_runtime.h>
#include <stdint.h>

// Problem constants (from reference: B=128, H=W=512, 3 stages x 15 iters)
#define BB 128
#define HH 512
#define WW 512
#define NITER 15
#define NSTAGES 3
#define TILE 64
#define HALO 15
#define REGION (TILE + 2*HALO)   // 94 rows/cols strictly required
#define LW 96                    // padded LDS tile dim (float4-aligned rows)
#define LSZ (LW*LW)              // 9216 floats per LDS plane

// ---- feature detection -----------------------------------------------------
#if defined(__has_builtin)
#if __has_builtin(__builtin_amdgcn_tensor_load_to_lds) && __has_builtin(__builtin_amdgcn_s_wait_tensorcnt)
#define HAVE_TDM 1
#endif
#endif
#ifndef HAVE_TDM
#define HAVE_TDM 0
#endif

// ---- CDNA5 async global->LDS copy (gfx1250 ASYNCcnt path) ------------------
__device__ __forceinline__ unsigned lds_off_of(const void* p) {
  // low 32 bits of a flat LDS address == workgroup-relative LDS byte offset
  return (unsigned)(uintptr_t)p;
}

__device__ __forceinline__ void async_copy_b32(const float* gsrc, unsigned lds_byte_off) {
  // VDST = per-lane LDS byte offset, VADDR = 64-bit global address, no SADDR
  asm volatile("global_load_async_to_lds_b32 %0, %1, off"
               :: "v"(lds_byte_off), "v"((unsigned long long)(uintptr_t)gsrc)
               : "memory");
}

__device__ __forceinline__ void wait_async_zero() {
  asm volatile("s_wait_asynccnt 0" ::: "memory");
}

// ---- CDNA5 Tensor Data Mover: one-shot 96x96 f32 tile -> LDS ---------------
#if HAVE_TDM
typedef unsigned int v4u __attribute__((ext_vector_type(4)));
typedef int          v8i __attribute__((ext_vector_type(8)));
typedef int          v4i __attribute__((ext_vector_type(4)));

__device__ __forceinline__ void tdm_load_tile_96x96(const float* gsrc, unsigned lds_byte,
                                                    unsigned row_stride_elems) {
  // D# per CDNA5 ISA 8.3-8.6: 2D tile, data_size=4B, tile 96x96, contiguous LDS
  unsigned long long ga = (unsigned long long)(uintptr_t)gsrc;
  v4u g0;
  g0[0] = 1u;                                          // count=1 (valid), user mode
  g0[1] = lds_byte;                                    // lds_addr
  g0[2] = (unsigned)(ga & 0xFFFFFFFFu);                // global_addr[31:0]
  g0[3] = (unsigned)((ga >> 32) & 0x01FFFFFFu)         // global_addr[56:32]
          | (2u << 30);                                // type=2 ("image")
  v8i g1;
  g1[0] = (int)(2u << 16);                             // data_size=4B; no mask/iter/pad
  g1[1] = (int)(96u << 16);                            // tensor_dim0[15:0]=96 @bits[63:48]
  g1[2] = (int)(96u << 16);                            // tensor_dim1[15:0]=96 @bits[95:80]
  g1[3] = (int)(96u << 16);                            // tile_dim0=96 @bits[127:112]
  g1[4] = 96;                                          // tile_dim1=96, tile_dim2=0
  g1[5] = (int)row_stride_elems;                       // tensor_dim0_stride (elements)
  g1[6] = 0;
  g1[7] = 0;                                           // tensor_dim1_stride unused
  v4i g2; g2[0] = 1; g2[1] = 1; g2[2] = 0; g2[3] = 0;  // dims2/3 unused (tile_dim3=0)
  v4i g3; g3[0] = 0; g3[1] = 0; g3[2] = 0; g3[3] = 0;
#if __clang_major__ >= 23
  v8i gz; gz[0]=0; gz[1]=0; gz[2]=0; gz[3]=0; gz[4]=0; gz[5]=0; gz[6]=0; gz[7]=0;
  __builtin_amdgcn_tensor_load_to_lds(g0, g1, g2, g3, gz, 0);
#else
  __builtin_amdgcn_tensor_load_to_lds(g0, g1, g2, g3, 0);
#endif
}
#endif

// ---- helpers ---------------------------------------------------------------
__global__ void zero_f32(float* p, size_t n4) {
  size_t i = (size_t)blockIdx.x * blockDim.x + threadIdx.x;
  size_t stride = (size_t)gridDim.x * blockDim.x;
  float4* v = (float4*)p;
  for (; i < n4; i += stride) v[i] = make_float4(0.f, 0.f, 0.f, 0.f);
}

__global__ void zero_keys(float* keys) {
  if (threadIdx.x < BB) keys[threadIdx.x] = 0.f;
}

__global__ void seed_kernel(float* R, const int* __restrict__ start) {
  int b = threadIdx.x;
  if (b < BB) {
    int r = min(max(start[2*b + 0], 0), HH - 1);
    int c = min(max(start[2*b + 1], 0), WW - 1);
    R[(size_t)b * HH * WW + (size_t)r * WW + c] = 1.0f;
  }
}

// ---- temporal-blocked stencil: 15 Jacobi iterations entirely in LDS --------
__global__ void __launch_bounds__(256)
stencil_stage(const float* __restrict__ Rin, float* __restrict__ Rout,
              const float* __restrict__ floorp, const float* __restrict__ doorp,
              const float* __restrict__ keys, const float* __restrict__ key_gate)
{
  __shared__ __align__(16) float smem[3 * LSZ];   // 108 KB: cur | nxt | walk
  float* cur  = smem;
  float* nxt  = smem + LSZ;
  float* walk = smem + 2 * LSZ;

  const int tid = threadIdx.x;
  const int tx = blockIdx.x, ty = blockIdx.y, b = blockIdx.z;
  const int gr0 = ty * TILE - HALO;
  const int gc0 = tx * TILE - HALO;
  const size_t plane = (size_t)HH * WW;
  const float* Rb = Rin    + (size_t)b * plane;
  const float* Fb = floorp + (size_t)b * plane;
  const float* Db = doorp  + (size_t)b * plane;

  // per-batch door pass scalar: sigmoid(kg * (keys[b] - 1))
  const float kg = key_gate[0];
  const float dp = 1.0f / (1.0f + expf(-kg * (keys[b] - 1.0f)));

  const unsigned curB = lds_off_of(cur);
  const unsigned nxtB = lds_off_of(nxt);
  const unsigned wkB  = lds_off_of(walk);

  const bool interior = (gr0 >= 0) && (gc0 >= 0) && (gr0 + LW <= HH) && (gc0 + LW <= WW);
#if HAVE_TDM
  if (interior) {
    // TDM overwrites all three full 96x96 planes -> no LDS pre-zero needed.
    // 3 descriptor-driven DMA tile loads, issued once (wave 0), TENSORcnt wait.
    if (tid == 0) {
      const size_t g = (size_t)gr0 * WW + gc0;
      tdm_load_tile_96x96(Rb + g, curB, WW);
      tdm_load_tile_96x96(Fb + g, wkB,  WW);
      tdm_load_tile_96x96(Db + g, nxtB, WW);
      __builtin_amdgcn_s_wait_tensorcnt(0);
    }
  } else
#endif
  {
    // Boundary tiles (or no-TDM fallback): zero planes so out-of-domain cells
    // stay the zero boundary, then per-lane async fill of the clamped region.
    {
      float4* sv = (float4*)smem;
      const float4 z = make_float4(0.f, 0.f, 0.f, 0.f);
      for (int i = tid; i < (3 * LSZ) / 4; i += 256) sv[i] = z;
    }
    __syncthreads();
    // R -> cur, floor -> walk(raw), door -> nxt(temp)
    for (int i = tid; i < REGION * REGION; i += 256) {
      int lr = i / REGION, lc = i - lr * REGION;
      int gr = gr0 + lr,  gc = gc0 + lc;
      if ((unsigned)gr < (unsigned)HH && (unsigned)gc < (unsigned)WW) {
        size_t g = (size_t)gr * WW + gc;
        unsigned lo = (unsigned)(lr * LW + lc) * 4u;
        async_copy_b32(Rb + g, curB + lo);
        async_copy_b32(Fb + g, wkB + lo);
        async_copy_b32(Db + g, nxtB + lo);
      }
    }
    wait_async_zero();
  }
  __syncthreads();

  // walk = floor * (1 - d + d*dp) * 0.25   (stencil weight folded in)
  for (int i = tid; i < LSZ; i += 256) {
    float d = nxt[i];
    walk[i] = walk[i] * (1.0f - d + d * dp) * 0.25f;
  }
  __syncthreads();

  // 15 temporal iterations, float4-vectorized over the 96x96 region
  for (int it = 0; it < NITER; ++it) {
    for (int q = tid; q < LSZ / 4; q += 256) {
      int row = q / (LW / 4);
      int c0  = (q - row * (LW / 4)) * 4;
      const float* rowp = cur + row * LW;
      const float4 c = *(const float4*)(rowp + c0);
      float left  = (c0 > 0)       ? rowp[c0 - 1] : 0.0f;
      float right = (c0 + 4 < LW)  ? rowp[c0 + 4] : 0.0f;
      const float4 z = make_float4(0.f, 0.f, 0.f, 0.f);
      float4 up = (row > 0)      ? *(const float4*)(cur + (row - 1) * LW + c0) : z;
      float4 dn = (row + 1 < LW) ? *(const float4*)(cur + (row + 1) * LW + c0) : z;
      const float4 wk = *(const float4*)(walk + row * LW + c0);
      float4 t, o;
      t.x = (up.x + dn.x) + (left + c.y);
      t.y = (up.y + dn.y) + (c.x + c.z);
      t.z = (up.z + dn.z) + (c.y + c.w);
      t.w = (up.w + dn.w) + (c.z + right);
      o.x = fminf(fmaxf(fmaf(t.x, wk.x, c.x), 0.f), 1.f);
      o.y = fminf(fmaxf(fmaf(t.y, wk.y, c.y), 0.f), 1.f);
      o.z = fminf(fmaxf(fmaf(t.z, wk.z, c.z), 0.f), 1.f);
      o.w = fminf(fmaxf(fmaf(t.w, wk.w, c.w), 0.f), 1.f);
      *(float4*)(nxt + row * LW + c0) = o;
    }
    __syncthreads();
    float* tmp = cur; cur = nxt; nxt = tmp;
  }

  // write back the 64x64 interior (disjoint across blocks; ping-pong buffer)
  float* Ro = Rout + (size_t)b * plane;
  for (int k = 0; k < 4; ++k) {
    int qid = tid + k * 256;         // 0..1023
    int row = qid >> 4;              // 0..63
    int qc  = qid & 15;              // 0..15
    const float* rowp = cur + (HALO + row) * LW + (HALO + qc * 4);
    float4 o;
    o.x = rowp[0]; o.y = rowp[1]; o.z = rowp[2]; o.w = rowp[3];
    *(float4*)(Ro + (size_t)(ty * TILE + row) * WW + (tx * TILE + qc * 4)) = o;
  }
}

// ---- deterministic two-pass reduction: keys[b] += sum(R * key_loc) ---------
__global__ void __launch_bounds__(256)
key_partial(const float* __restrict__ R, const float* __restrict__ keyloc,
            float* __restrict__ partials)
{
  const int b = blockIdx.y, chunk = blockIdx.x, tid = threadIdx.x;
  const size_t base = (size_t)b * HH * WW + (size_t)chunk * 1024;
  const float4 r = ((const float4*)(R + base))[tid];
  const float4 k = ((const float4*)(keyloc + base))[tid];
  float s = r.x * k.x + r.y * k.y + r.z * k.z + r.w * k.w;
  __shared__ float red[256];
  red[tid] = s; __syncthreads();
  for (int off = 128; off > 0; off >>= 1) {
    if (tid < off) red[tid] += red[tid + off];
    __syncthreads();
  }
  if (tid == 0) partials[b * 256 + chunk] = red[0];
}

__global__ void __launch_bounds__(256)
key_reduce(const float* __restrict__ partials, float* __restrict__ keys)
{
  const int b = blockIdx.x, tid = threadIdx.x;
  __shared__ float red[256];
  red[tid] = partials[b * 256 + tid]; __syncthreads();
  for (int off = 128; off > 0; off >>= 1) {
    if (tid < off) red[tid] += red[tid + off];
    __syncthreads();
  }
  if (tid == 0) keys[b] += red[0];
}

__global__ void gather_goal(const float* __restrict__ R, const int* __restrict__ goal,
                            float* __restrict__ out)
{
  int b = threadIdx.x;
  if (b < BB) {
    int r = min(max(goal[2*b + 0], 0), HH - 1);
    int c = min(max(goal[2*b + 1], 0), WW - 1);
    out[b] = R[(size_t)b * HH * WW + (size_t)r * WW + c];
  }
}

// ---------------------------------------------------------------------------
extern "C" void kernel_launch(void* const* d_in, const int* in_sizes, int n_in,
                              void* d_out, int out_size, void* d_ws, size_t ws_size,
                              hipStream_t stream) {
  (void)in_sizes; (void)n_in; (void)out_size; (void)ws_size;
  const float* floorp = (const float*)d_in[0];
  const float* keyloc = (const float*)d_in[1];
  const float* doorp  = (const float*)d_in[2];
  const int*   start  = (const int*)d_in[3];
  const int*   goal   = (const int*)d_in[4];
  const float* kg     = (const float*)d_in[5];

  char* ws = (char*)d_ws;
  const size_t plane_bytes = (size_t)BB * HH * WW * sizeof(float);   // 128 MiB
  float* bufA     = (float*)(ws);
  float* bufB     = (float*)(ws + plane_bytes);
  float* keys     = (float*)(ws + 2 * plane_bytes);
  float* partials = (float*)(ws + 2 * plane_bytes + 4096);

  zero_f32<<<2048, 256, 0, stream>>>(bufA, (size_t)BB * HH * WW / 4);
  zero_keys<<<1, 128, 0, stream>>>(keys);
  seed_kernel<<<1, 128, 0, stream>>>(bufA, start);

  float* bufs[2] = { bufA, bufB };
  dim3 sgrid(WW / TILE, HH / TILE, BB);   // 8 x 8 x 128 tiles
  dim3 rgrid(256, BB);
  for (int s = 0; s < NSTAGES; ++s) {
    float* rin  = bufs[s & 1];
    float* rout = bufs[(s + 1) & 1];
    stencil_stage<<<sgrid, 256, 0, stream>>>(rin, rout, floorp, doorp, keys, kg);
    key_partial<<<rgrid, 256, 0, stream>>>(rout, keyloc, partials);
    key_reduce<<<BB, 256, 0, stream>>>(partials, keys);
  }
  gather_goal<<<1, 128, 0, stream>>>(bufs[NSTAGES & 1], goal, (float*)d_out);
}